// GAT_85899346613
// MI455X (gfx1250) — compile-verified
//
#include <hip/hip_runtime.h>
#include <hip/hip_bf16.h>
#include <limits.h>

// ---------------------------------------------------------------------------
// GAT 2-layer forward for MI455X (gfx1250, wave32).
//  - Dense GEMM h1 = x @ W1 uses V_WMMA_F32_16X16X4_F32 (full fp32 matrix
//    pipe), with W1 staged through LDS (bank-conflict-padded) and 2-way
//    M register blocking per wave.
//  - Edge softmax/scatter phases are memory/atomic bound: VALU + f32 atomics.
// Problem constants from the reference: F_in=128, H=8, C=16, OUT=2.
// ---------------------------------------------------------------------------

typedef __attribute__((ext_vector_type(2))) float v2f;
typedef __attribute__((ext_vector_type(8))) float v8f;

#define F_IN   128
#define HEADS  8
#define CH     16
#define OUTC   2
#define NEG_SLOPE 0.2f
#define LDSW   136   // 128 + 8 pad: B-fragment halves hit disjoint bank sets

// ---- ordered-int encoding for float atomic max (handles negatives) --------
__device__ __forceinline__ int f2ord(float f) {
  int i = __float_as_int(f);
  return (i >= 0) ? i : (i ^ 0x7fffffff);
}
__device__ __forceinline__ float ord2f(int i) {
  return __int_as_float((i >= 0) ? i : (i ^ 0x7fffffff));
}

// ---- edge endpoints: first E entries from edge_index, then N self loops ---
__device__ __forceinline__ void edge_sd(const int* __restrict__ ei, int E,
                                        int e, int& s, int& d) {
  if (e < E) { s = ei[e]; d = ei[E + e]; }
  else       { s = e - E; d = e - E; }
}

// ---------------------------------------------------------------------------
// fills
// ---------------------------------------------------------------------------
__global__ void gat_fill_f32(float* __restrict__ p, float v, size_t n) {
  size_t i = (size_t)blockIdx.x * blockDim.x + threadIdx.x;
  if (i < n) p[i] = v;
}
__global__ void gat_fill_i32(int* __restrict__ p, int v, size_t n) {
  size_t i = (size_t)blockIdx.x * blockDim.x + threadIdx.x;
  if (i < n) p[i] = v;
}

// ---------------------------------------------------------------------------
// Kernel 1: h1[N,128] = x[N,128] @ W1[128,128] via WMMA F32 16x16x4.
// Block = 8 waves, tile 32 rows x 128 cols (N % 32 == 0 -> no remainder).
// Wave w owns cols [16w,16w+16) and TWO 16x16 M-tiles (2 accumulators,
// shared B fragment). W1 staged through LDS in 32-row K-chunks.
//
// Fragment layouts (ISA 7.12.2, 32-bit):
//  A 16x4 : lanes 0-15 M=lane K={0,1}; lanes 16-31 M=lane-16 K={2,3}
//  B 4x16 : lanes 0-15 N=lane K={0,1}; lanes 16-31 N=lane-16 K={2,3}
//  C/D    : VGPR r holds rows M=r (lanes 0-15) and M=8+r (lanes 16-31)
// ---------------------------------------------------------------------------
__global__ __launch_bounds__(256) void gat_gemm_wmma_f32(
    const float* __restrict__ X, const float* __restrict__ W,
    float* __restrict__ H1, int N) {
  __shared__ float Wl[32 * LDSW];      // 17 KB, padded rows

  const int t    = threadIdx.x;
  const int wave = t >> 5;             // 0..7 -> 16-col block
  const int lane = t & 31;
  const int half = lane >> 4;          // 0: K={0,1}   1: K={2,3}
  const int l    = lane & 15;          // M index (A) / N index (B)
  const int row0 = blockIdx.x * 32;
  const int col0 = wave * 16;

  const float* __restrict__ xr0 = X + (size_t)(row0 + l) * F_IN;
  const float* __restrict__ xr1 = X + (size_t)(row0 + 16 + l) * F_IN;
  v8f acc0 = {}, acc1 = {};

  for (int kc = 0; kc < F_IN; kc += 32) {
    // stage W1 rows [kc, kc+32): 4096 floats, 4 x float4 per thread
    for (int i = t * 4; i < 32 * F_IN; i += 256 * 4) {
      const float4 v = *(const float4*)(W + (size_t)kc * F_IN + i);
      float* p = &Wl[(i >> 7) * LDSW + (i & 127)];
      p[0] = v.x; p[1] = v.y; p[2] = v.z; p[3] = v.w;
    }
    __syncthreads();

#pragma unroll
    for (int k = 0; k < 32; k += 4) {
      v2f b;
      b.x = Wl[(k + 2 * half)     * LDSW + col0 + l];
      b.y = Wl[(k + 2 * half + 1) * LDSW + col0 + l];
      const int ka = kc + k + 2 * half;
      v2f a0, a1;
      a0.x = xr0[ka]; a0.y = xr0[ka + 1];
      a1.x = xr1[ka]; a1.y = xr1[ka + 1];
      acc0 = __builtin_amdgcn_wmma_f32_16x16x4_f32(
          false, a0, false, b, (short)0, acc0, false, false);
      acc1 = __builtin_amdgcn_wmma_f32_16x16x4_f32(
          false, a1, false, b, (short)0, acc1, false, false);
    }
    __syncthreads();
  }

  float* __restrict__ out0 = H1 + (size_t)row0 * F_IN + col0;
  float* __restrict__ out1 = H1 + (size_t)(row0 + 16) * F_IN + col0;
#pragma unroll
  for (int r = 0; r < 8; ++r) {
    out0[(size_t)(r + 8 * half) * F_IN + l] = acc0[r];
    out1[(size_t)(r + 8 * half) * F_IN + l] = acc1[r];
  }
}

// ---------------------------------------------------------------------------
// Kernel 2: per-(node,head) attention dots  a_src1/a_dst1 [N,8]
// ---------------------------------------------------------------------------
__global__ __launch_bounds__(256) void gat_attn1(
    const float* __restrict__ H1, const float* __restrict__ att_src,
    const float* __restrict__ att_dst, float* __restrict__ as1,
    float* __restrict__ ad1, int N) {
  int idx = blockIdx.x * blockDim.x + threadIdx.x;
  if (idx >= N * HEADS) return;
  int n = idx >> 3, h = idx & 7;
  const float* __restrict__ hp = H1 + (size_t)n * F_IN + h * CH;
  float s = 0.f, d = 0.f;
#pragma unroll
  for (int c = 0; c < CH; ++c) {
    float v = hp[c];
    s = fmaf(v, att_src[h * CH + c], s);
    d = fmaf(v, att_dst[h * CH + c], d);
  }
  as1[idx] = s;
  ad1[idx] = d;
}

// ---------------------------------------------------------------------------
// Edge pass A: e = leaky_relu(a_src[s] + a_dst[d]); segment max via
// ordered-int atomic max. One thread per (edge, head).
// ---------------------------------------------------------------------------
__global__ __launch_bounds__(256) void gat_edge_max(
    const int* __restrict__ ei, const float* __restrict__ as,
    const float* __restrict__ ad, float* __restrict__ eb,
    int* __restrict__ m, int E, int Et, int H) {
  long long idx = (long long)blockIdx.x * blockDim.x + threadIdx.x;
  if (idx >= (long long)Et * H) return;
  int e = (int)(idx / H), h = (int)(idx % H);
  int s, d; edge_sd(ei, E, e, s, d);
  float v = as[s * H + h] + ad[d * H + h];
  v = (v > 0.f) ? v : NEG_SLOPE * v;
  eb[idx] = v;
  atomicMax(&m[d * H + h], f2ord(v));
}

// ---------------------------------------------------------------------------
// Edge pass B: ex = exp(e - m[d]); segment sum via f32 atomic add.
// ---------------------------------------------------------------------------
__global__ __launch_bounds__(256) void gat_edge_exp(
    const int* __restrict__ ei, float* __restrict__ eb,
    const int* __restrict__ m, float* __restrict__ ssum,
    int E, int Et, int H) {
  long long idx = (long long)blockIdx.x * blockDim.x + threadIdx.x;
  if (idx >= (long long)Et * H) return;
  int e = (int)(idx / H), h = (int)(idx % H);
  int s, d; edge_sd(ei, E, e, s, d);
  (void)s;
  float ex = __expf(eb[idx] - ord2f(m[d * H + h]));
  eb[idx] = ex;
  atomicAdd(&ssum[d * H + h], ex);
}

// ---------------------------------------------------------------------------
// Edge pass C (layer 1): scatter alpha * h[src] into out[dst].
// One thread per (edge, channel) -> 128 threads/edge, coalesced h1 reads.
// ---------------------------------------------------------------------------
__global__ __launch_bounds__(256) void gat_edge_scatter1(
    const int* __restrict__ ei, const float* __restrict__ eb,
    const float* __restrict__ ssum, const float* __restrict__ H1,
    float* __restrict__ O1, int E, int Et) {
  long long idx = (long long)blockIdx.x * blockDim.x + threadIdx.x;
  if (idx >= (long long)Et * F_IN) return;
  int e = (int)(idx >> 7), c = (int)(idx & 127), h = c >> 4;
  int s, d; edge_sd(ei, E, e, s, d);
  float alpha = eb[(size_t)e * HEADS + h] / (ssum[d * HEADS + h] + 1e-16f);
  atomicAdd(&O1[(size_t)d * F_IN + c], alpha * H1[(size_t)s * F_IN + c]);
}

// ---------------------------------------------------------------------------
// Bias + ELU. Writes into the (now dead) h1 buffer -> hact.
// ---------------------------------------------------------------------------
__global__ __launch_bounds__(256) void gat_bias_elu(
    const float* __restrict__ O1, const float* __restrict__ b1,
    float* __restrict__ HACT, int N) {
  long long idx = (long long)blockIdx.x * blockDim.x + threadIdx.x;
  if (idx >= (long long)N * F_IN) return;
  int c = (int)(idx & 127);
  float v = O1[idx] + b1[c];
  HACT[idx] = (v > 0.f) ? v : (__expf(v) - 1.f);
}

// ---------------------------------------------------------------------------
// Layer-2 node kernel: h2[N,2] = hact @ W2[128,2]; attention dots (1 head).
// ---------------------------------------------------------------------------
__global__ __launch_bounds__(256) void gat_layer2_node(
    const float* __restrict__ HACT, const float* __restrict__ W2,
    const float* __restrict__ att_src2, const float* __restrict__ att_dst2,
    float* __restrict__ H2, float* __restrict__ as2, float* __restrict__ ad2,
    int N) {
  int n = blockIdx.x * blockDim.x + threadIdx.x;
  if (n >= N) return;
  const float* __restrict__ hp = HACT + (size_t)n * F_IN;
  float a0 = 0.f, a1 = 0.f;
#pragma unroll 8
  for (int k = 0; k < F_IN; ++k) {
    float v = hp[k];
    a0 = fmaf(v, W2[k * OUTC + 0], a0);
    a1 = fmaf(v, W2[k * OUTC + 1], a1);
  }
  H2[n * OUTC + 0] = a0;
  H2[n * OUTC + 1] = a1;
  as2[n] = a0 * att_src2[0] + a1 * att_src2[1];
  ad2[n] = a0 * att_dst2[0] + a1 * att_dst2[1];
}

// ---------------------------------------------------------------------------
// Edge pass C (layer 2): scatter alpha * h2[src] into o2[dst]. 2 thr/edge.
// ---------------------------------------------------------------------------
__global__ __launch_bounds__(256) void gat_edge_scatter2(
    const int* __restrict__ ei, const float* __restrict__ eb,
    const float* __restrict__ ssum, const float* __restrict__ H2,
    float* __restrict__ O2, int E, int Et) {
  long long idx = (long long)blockIdx.x * blockDim.x + threadIdx.x;
  if (idx >= (long long)Et * OUTC) return;
  int e = (int)(idx >> 1), j = (int)(idx & 1);
  int s, d; edge_sd(ei, E, e, s, d);
  float alpha = eb[e] / (ssum[d] + 1e-16f);
  atomicAdd(&O2[(size_t)d * OUTC + j], alpha * H2[(size_t)s * OUTC + j]);
}

// ---------------------------------------------------------------------------
// Final: out = o2 + b2   (heads=1, concat=False -> mean over 1 head = id)
// ---------------------------------------------------------------------------
__global__ __launch_bounds__(256) void gat_final(
    const float* __restrict__ O2, const float* __restrict__ b2,
    float* __restrict__ OUT, int N) {
  int idx = blockIdx.x * blockDim.x + threadIdx.x;
  if (idx >= N * OUTC) return;
  OUT[idx] = O2[idx] + b2[idx & 1];
}

// ---------------------------------------------------------------------------
// launch
// ---------------------------------------------------------------------------
static inline unsigned gdim(long long n, int b) {
  return (unsigned)((n + b - 1) / b);
}

extern "C" void kernel_launch(void* const* d_in, const int* in_sizes, int n_in,
                              void* d_out, int out_size, void* d_ws,
                              size_t ws_size, hipStream_t stream) {
  const float* x        = (const float*)d_in[0];
  const int*   ei       = (const int*)d_in[1];   // [2,E] flat: src row then dst row
  const float* W1       = (const float*)d_in[2];
  const float* att_src1 = (const float*)d_in[3];
  const float* att_dst1 = (const float*)d_in[4];
  const float* b1       = (const float*)d_in[5];
  const float* W2       = (const float*)d_in[6];
  const float* att_src2 = (const float*)d_in[7];
  const float* att_dst2 = (const float*)d_in[8];
  const float* b2       = (const float*)d_in[9];

  const int N  = in_sizes[0] / F_IN;   // 100000
  const int E  = in_sizes[1] / 2;      // 800000
  const int Et = E + N;                // edges incl. self loops

  // workspace carve-up (fp32 elements)
  float* ws = (float*)d_ws;
  size_t off = 0;
  float* h1  = ws + off; off += (size_t)N * F_IN;   // reused as hact after scatter1
  float* as1 = ws + off; off += (size_t)N * HEADS;
  float* ad1 = ws + off; off += (size_t)N * HEADS;
  float* e1  = ws + off; off += (size_t)Et * HEADS;
  int*   m1  = (int*)(ws + off); off += (size_t)N * HEADS;
  float* s1  = ws + off; off += (size_t)N * HEADS;
  float* o1  = ws + off; off += (size_t)N * F_IN;
  float* h2  = ws + off; off += (size_t)N * OUTC;
  float* as2 = ws + off; off += (size_t)N;
  float* ad2 = ws + off; off += (size_t)N;
  float* e2  = ws + off; off += (size_t)Et;
  int*   m2  = (int*)(ws + off); off += (size_t)N;
  float* s2  = ws + off; off += (size_t)N;
  float* o2  = ws + off; off += (size_t)N * OUTC;
  (void)ws_size; (void)n_in; (void)out_size;

  const int B = 256;

  // zero / -inf init of all accumulators (required every call)
  gat_fill_i32<<<gdim((long long)N * HEADS, B), B, 0, stream>>>(m1, INT_MIN, (size_t)N * HEADS);
  gat_fill_f32<<<gdim((long long)N * HEADS, B), B, 0, stream>>>(s1, 0.f, (size_t)N * HEADS);
  gat_fill_f32<<<gdim((long long)N * F_IN, B), B, 0, stream>>>(o1, 0.f, (size_t)N * F_IN);
  gat_fill_i32<<<gdim(N, B), B, 0, stream>>>(m2, INT_MIN, (size_t)N);
  gat_fill_f32<<<gdim(N, B), B, 0, stream>>>(s2, 0.f, (size_t)N);
  gat_fill_f32<<<gdim((long long)N * OUTC, B), B, 0, stream>>>(o2, 0.f, (size_t)N * OUTC);

  // layer 1
  gat_gemm_wmma_f32<<<gdim(N, 32), 256, 0, stream>>>(x, W1, h1, N);
  gat_attn1<<<gdim((long long)N * HEADS, B), B, 0, stream>>>(h1, att_src1, att_dst1, as1, ad1, N);
  gat_edge_max<<<gdim((long long)Et * HEADS, B), B, 0, stream>>>(ei, as1, ad1, e1, m1, E, Et, HEADS);
  gat_edge_exp<<<gdim((long long)Et * HEADS, B), B, 0, stream>>>(ei, e1, m1, s1, E, Et, HEADS);
  gat_edge_scatter1<<<gdim((long long)Et * F_IN, B), B, 0, stream>>>(ei, e1, s1, h1, o1, E, Et);
  gat_bias_elu<<<gdim((long long)N * F_IN, B), B, 0, stream>>>(o1, b1, h1, N);  // h1 := hact

  // layer 2
  gat_layer2_node<<<gdim(N, B), B, 0, stream>>>(h1, W2, att_src2, att_dst2, h2, as2, ad2, N);
  gat_edge_max<<<gdim((long long)Et, B), B, 0, stream>>>(ei, as2, ad2, e2, m2, E, Et, 1);
  gat_edge_exp<<<gdim((long long)Et, B), B, 0, stream>>>(ei, e2, m2, s2, E, Et, 1);
  gat_edge_scatter2<<<gdim((long long)Et * OUTC, B), B, 0, stream>>>(ei, e2, s2, h2, o2, E, Et);
  gat_final<<<gdim((long long)N * OUTC, B), B, 0, stream>>>(o2, b2, (float*)d_out, N);
}